// ODEFunc_interaction_12378095747592
// MI455X (gfx1250) — compile-verified
//
#include <hip/hip_runtime.h>
#include <math.h>

typedef float v2f __attribute__((ext_vector_type(2)));
typedef float v8f __attribute__((ext_vector_type(8)));

static __device__ __forceinline__ v8f wmma4(v2f a, v2f b, v8f c) {
  // D = A(16x4,f32) * B(4x16,f32) + C(16x16,f32)
  return __builtin_amdgcn_wmma_f32_16x16x4_f32(false, a, false, b, (short)0, c,
                                               false, false);
}

static __device__ __forceinline__ float fast_tanh(float x) {
#if __has_builtin(__builtin_amdgcn_tanhf)
  return __builtin_amdgcn_tanhf(x);  // V_TANH_F32 (gfx1250 transcendental)
#else
  return tanhf(x);
#endif
}

#define HSTRIDE 132     // row stride (floats) for 16x128 staging: banks 4m+k
#define W1P_STRIDE 288  // k-pair stride for 128-wide B: 256 data + 32 pad
#define W2P_STRIDE 96   // k-pair stride for 32-wide B: 64 data + 32 pad

// ---------------------------------------------------------------------------
// Kernel 1: out = tanh([z | t] @ vW1 + vb1) @ vW2 + vb2
// One wave = 16 rows; 8 waves/block = 128 rows/block.
// B-weights stored pair-interleaved: frag = single conflict-free ds_load_b64.
// ---------------------------------------------------------------------------
__global__ __launch_bounds__(256) void mlp_kernel(
    const float* __restrict__ z, const float* __restrict__ t,
    const float* __restrict__ vW1, const float* __restrict__ vb1,
    const float* __restrict__ vW2, const float* __restrict__ vb2,
    float* __restrict__ out, int B) {
  __shared__ float sW1p[18 * W1P_STRIDE];  // [k/2][2n + (k&1)], k padded to 36
  __shared__ float sW2p[64 * W2P_STRIDE];  // [k/2][2n + (k&1)]
  __shared__ float sH[8 * 16 * HSTRIDE];   // per-wave h staging

  const int tid = threadIdx.x;
  for (int i = tid; i < 36 * 128; i += 256) {
    const int k = i >> 7, n = i & 127;
    sW1p[(k >> 1) * W1P_STRIDE + 2 * n + (k & 1)] =
        (i < 33 * 128) ? vW1[i] : 0.0f;
  }
  for (int i = tid; i < 128 * 32; i += 256) {
    const int k = i >> 5, n = i & 31;
    sW2p[(k >> 1) * W2P_STRIDE + 2 * n + (k & 1)] = vW2[i];
  }
  __syncthreads();

  const int wave = tid >> 5;
  const int lane = tid & 31;
  const int laneM = lane & 15;
  const int laneHi = lane >> 4;
  const int k0off = 2 * laneHi;

  const long long rowBase = (long long)blockIdx.x * 128 + wave * 16;
  const long long m = rowBase + laneM;
  // clamp instead of predicating loads: garbage rows are masked at the store
  const long long mc = (m < (long long)B) ? m : (long long)(B - 1);
  const float* __restrict__ zrow = z + mc * 32;
  const float tval = t[0];

  v8f acc[8];
#pragma unroll
  for (int j = 0; j < 8; ++j) acc[j] = v8f{};

  // ---- layer 1: X(16x36) @ vW1(36x128) ----
#pragma unroll
  for (int kt = 0; kt < 8; ++kt) {
    const v2f a = *(const v2f*)(zrow + kt * 4 + k0off);
    const int k2 = 2 * kt + laneHi;
#pragma unroll
    for (int j = 0; j < 8; ++j) {
      const v2f b =
          *(const v2f*)(&sW1p[k2 * W1P_STRIDE + 2 * (j * 16 + laneM)]);
      acc[j] = wmma4(a, b, acc[j]);
    }
  }
  {  // k-tile 8: column 32 is t, 33..35 are zero
    v2f a = {0.0f, 0.0f};
    if (laneHi == 0) a.x = tval;
    const int k2 = 16 + laneHi;
#pragma unroll
    for (int j = 0; j < 8; ++j) {
      const v2f b =
          *(const v2f*)(&sW1p[k2 * W1P_STRIDE + 2 * (j * 16 + laneM)]);
      acc[j] = wmma4(a, b, acc[j]);
    }
  }

  // bias + tanh (V_TANH_F32), stage h into LDS
  float* myH = &sH[wave * (16 * HSTRIDE)];
#pragma unroll
  for (int j = 0; j < 8; ++j) {
    const int n = j * 16 + laneM;
    const float b1 = vb1[n];
#pragma unroll
    for (int v = 0; v < 8; ++v) {
      myH[(v + 8 * laneHi) * HSTRIDE + n] = fast_tanh(acc[j][v] + b1);
    }
  }
  __syncthreads();

  // ---- layer 2: h(16x128) @ vW2(128x32) ----
  v8f d0 = v8f{}, d1 = v8f{};
#pragma unroll
  for (int kt = 0; kt < 32; ++kt) {
    const int k0 = kt * 4 + k0off;
    const int k2 = 2 * kt + laneHi;
    const v2f a = *(const v2f*)(&myH[laneM * HSTRIDE + k0]);
    const v2f b0 = *(const v2f*)(&sW2p[k2 * W2P_STRIDE + 2 * laneM]);
    d0 = wmma4(a, b0, d0);
    const v2f b1 = *(const v2f*)(&sW2p[k2 * W2P_STRIDE + 32 + 2 * laneM]);
    d1 = wmma4(a, b1, d1);
  }

  const float bias0 = vb2[laneM];
  const float bias1 = vb2[16 + laneM];
#pragma unroll
  for (int v = 0; v < 8; ++v) {
    const long long mr = rowBase + v + 8 * laneHi;
    if (mr < (long long)B) {
      out[mr * 32 + laneM] = d0[v] + bias0;
      out[mr * 32 + 16 + laneM] = d1[v] + bias1;
    }
  }
}

// ---------------------------------------------------------------------------
// Kernel 2: pair forces. d = z[perm[2p]] - z[perm[2p+1]],
// g = ((1 - tanh^2(d@pW1 + pb1)) * pW2) @ pW1^T ; out[i] -= g ; out[j] += g.
// perm is a permutation -> unique writer per element -> plain RMW is safe.
// ---------------------------------------------------------------------------
__global__ __launch_bounds__(256) void pair_kernel(
    const float* __restrict__ z, const int* __restrict__ perm,
    const float* __restrict__ pW1, const float* __restrict__ pb1,
    const float* __restrict__ pW2, float* __restrict__ out, int num_pairs) {
  __shared__ float sW1p[16 * W1P_STRIDE];  // pair-interleaved (layer-1 B)
  __shared__ float sW1r[32 * HSTRIDE];     // row-major padded (layer-2 B = pW1^T)
  __shared__ float sb1[128];
  __shared__ float sw2[128];
  __shared__ float sWm[8 * 16 * HSTRIDE];  // per-wave (1-u^2)*pW2 staging

  const int tid = threadIdx.x;
  for (int i = tid; i < 32 * 128; i += 256) {
    const int k = i >> 7, n = i & 127;
    const float v = pW1[i];
    sW1p[(k >> 1) * W1P_STRIDE + 2 * n + (k & 1)] = v;
    sW1r[k * HSTRIDE + n] = v;
  }
  if (tid < 128) {
    sb1[tid] = pb1[tid];
    sw2[tid] = pW2[tid];
  }
  __syncthreads();

  const int wave = tid >> 5;
  const int lane = tid & 31;
  const int laneM = lane & 15;
  const int laneHi = lane >> 4;
  const int k0off = 2 * laneHi;

  const long long pBase = (long long)blockIdx.x * 128 + wave * 16;
  const long long p = pBase + laneM;
  const bool pv = (p < (long long)num_pairs);
  const long long ip = pv ? 2 * p : 0;  // clamp: invalid pairs compute garbage,
  const int ri = perm[ip];              // masked at the scatter
  const int rj = perm[pv ? ip + 1 : 0];
  const float* __restrict__ zi = z + (long long)ri * 32;
  const float* __restrict__ zj = z + (long long)rj * 32;

  v8f acc[8];
#pragma unroll
  for (int j = 0; j < 8; ++j) acc[j] = v8f{};

  // ---- layer 1: D(16x32) @ pW1(32x128) ----
#pragma unroll
  for (int kt = 0; kt < 8; ++kt) {
    const int k0 = kt * 4 + k0off;
    const v2f a = *(const v2f*)(zi + k0) - *(const v2f*)(zj + k0);
    const int k2 = 2 * kt + laneHi;
#pragma unroll
    for (int j = 0; j < 8; ++j) {
      const v2f b =
          *(const v2f*)(&sW1p[k2 * W1P_STRIDE + 2 * (j * 16 + laneM)]);
      acc[j] = wmma4(a, b, acc[j]);
    }
  }

  // w = (1 - tanh^2(u + pb1)) * pW2 ; stage to LDS
  float* myW = &sWm[wave * (16 * HSTRIDE)];
#pragma unroll
  for (int j = 0; j < 8; ++j) {
    const int n = j * 16 + laneM;
    const float b1 = sb1[n];
    const float w2 = sw2[n];
#pragma unroll
    for (int v = 0; v < 8; ++v) {
      const float u = fast_tanh(acc[j][v] + b1);
      myW[(v + 8 * laneHi) * HSTRIDE + n] = (1.0f - u * u) * w2;
    }
  }
  __syncthreads();

  // ---- layer 2: G = W(16x128) @ pW1^T(128x32) ----
  v8f g0 = v8f{}, g1 = v8f{};
#pragma unroll
  for (int kt = 0; kt < 32; ++kt) {
    const int k0 = kt * 4 + k0off;
    const v2f a = *(const v2f*)(&myW[laneM * HSTRIDE + k0]);
    const v2f b0 = *(const v2f*)(&sW1r[laneM * HSTRIDE + k0]);
    g0 = wmma4(a, b0, g0);
    const v2f b1 = *(const v2f*)(&sW1r[(16 + laneM) * HSTRIDE + k0]);
    g1 = wmma4(a, b1, g1);
  }

  // scatter-add (unique writer per element)
#pragma unroll
  for (int v = 0; v < 8; ++v) {
    const long long pm = pBase + v + 8 * laneHi;
    if (pm < (long long)num_pairs) {
      const int im = perm[2 * pm];
      const int jm = perm[2 * pm + 1];
      const float f0 = g0[v];
      const float f1 = g1[v];
      out[(long long)im * 32 + laneM] -= f0;
      out[(long long)jm * 32 + laneM] += f0;
      out[(long long)im * 32 + 16 + laneM] -= f1;
      out[(long long)jm * 32 + 16 + laneM] += f1;
    }
  }
}

// ---------------------------------------------------------------------------
// Kernel 3: the single leftover triple (only when B is odd). One wave.
// force[a] = 2 * sum_{b != a} ( -grad_phi(z_a - z_b) )
// ---------------------------------------------------------------------------
__global__ void triple_kernel(const float* __restrict__ z,
                              const int* __restrict__ perm,
                              const float* __restrict__ pW1,
                              const float* __restrict__ pb1,
                              const float* __restrict__ pW2,
                              float* __restrict__ out, int pbase) {
  __shared__ float tz[3][32];
  __shared__ float w[128];
  const int lane = threadIdx.x;  // 32 threads
  int rows[3];
#pragma unroll
  for (int r = 0; r < 3; ++r) {
    rows[r] = perm[pbase + r];
    tz[r][lane] = z[(long long)rows[r] * 32 + lane];
  }
  __syncthreads();
  float facc[3] = {0.0f, 0.0f, 0.0f};
  for (int a = 0; a < 3; ++a) {
    for (int b = 0; b < 3; ++b) {
      if (a == b) continue;
#pragma unroll
      for (int hh = 0; hh < 4; ++hh) {
        const int h = lane + 32 * hh;
        float s = pb1[h];
        for (int d = 0; d < 32; ++d)
          s += (tz[a][d] - tz[b][d]) * pW1[d * 128 + h];
        const float u = fast_tanh(s);
        w[h] = (1.0f - u * u) * pW2[h];
      }
      __syncthreads();
      float g = 0.0f;
      for (int h = 0; h < 128; ++h) g += pW1[lane * 128 + h] * w[h];
      facc[a] -= 2.0f * g;
      __syncthreads();
    }
  }
#pragma unroll
  for (int a = 0; a < 3; ++a)
    out[(long long)rows[a] * 32 + lane] += facc[a];
}

// ---------------------------------------------------------------------------
extern "C" void kernel_launch(void* const* d_in, const int* in_sizes, int n_in,
                              void* d_out, int out_size, void* d_ws,
                              size_t ws_size, hipStream_t stream) {
  (void)n_in; (void)d_ws; (void)ws_size; (void)out_size;
  const float* t    = (const float*)d_in[0];
  const float* z    = (const float*)d_in[1];
  const int*   perm = (const int*)d_in[2];
  const float* vW1  = (const float*)d_in[3];
  const float* vb1  = (const float*)d_in[4];
  const float* vW2  = (const float*)d_in[5];
  const float* vb2  = (const float*)d_in[6];
  const float* pW1  = (const float*)d_in[7];
  const float* pb1  = (const float*)d_in[8];
  const float* pW2  = (const float*)d_in[9];
  // d_in[10] = pb2: constant shift, zero gradient -> unused
  float* out = (float*)d_out;

  const int B = in_sizes[2];  // perm length == batch size
  const int num_pairs = (B % 2 == 0) ? (B / 2) : ((B - 3) / 2);

  const int grid1 = (B + 127) / 128;
  mlp_kernel<<<grid1, 256, 0, stream>>>(z, t, vW1, vb1, vW2, vb2, out, B);

  if (num_pairs > 0) {
    const int grid2 = (num_pairs + 127) / 128;
    pair_kernel<<<grid2, 256, 0, stream>>>(z, perm, pW1, pb1, pW2, out,
                                           num_pairs);
  }
  if ((B % 2) == 1 && B >= 3) {
    triple_kernel<<<1, 32, 0, stream>>>(z, perm, pW1, pb1, pW2, out,
                                        2 * num_pairs);
  }
}